// StraddleLSTM_48060684043098
// MI455X (gfx1250) — compile-verified
//
#include <hip/hip_runtime.h>
#include <hip/hip_bf16.h>
#include <math.h>

// ---------------------------------------------------------------------------
// StraddleLSTM on MI455X (gfx1250), wave32, f32 WMMA 16x16x4.
//   B=512, S=2048, D=19, H=40, 4H=160
// Phase A (scan kernel): fused input-projection + LSTM recurrence,
//   software-pipelined x-tile prefetch, branchless HW transcendentals.
// Phase B (head kernel): LayerNorm + tanh-MLP + output head.
// ---------------------------------------------------------------------------

#define B_  512
#define S_  2048
#define D_  19
#define H_  40
#define G_  160          // 4H
#define LN_EPS 1e-5f

typedef __attribute__((ext_vector_type(2))) float v2f;
typedef __attribute__((ext_vector_type(8))) float v8f;

// output layout (floats): sig[B*S] | hN[B*H] | cN[B*H] | thr
#define SIG_N   ((size_t)B_ * S_)
#define HN_OFF  (SIG_N)
#define CN_OFF  (SIG_N + (size_t)B_ * H_)
#define THR_OFF (SIG_N + 2 * (size_t)B_ * H_)

// Branchless transcendentals: v_exp_f32 + v_rcp_f32, clamp keeps exp finite
// (tanh/sigmoid are saturated to 1 ulp beyond the clamp range anyway).
__device__ __forceinline__ float fast_sig(float v) {
    const float x = fminf(fmaxf(v, -30.0f), 30.0f);
    return __fdividef(1.0f, 1.0f + __expf(-x));
}
__device__ __forceinline__ float fast_tanh(float v) {
    const float x = fminf(fmaxf(v, -15.0f), 15.0f);
    const float t = __expf(2.0f * x);
    return __fdividef(t - 1.0f, t + 1.0f);
}

// D = A(16x4 f32) * B(4x16 f32) + C(16x16 f32)
__device__ __forceinline__ v8f wmma4(v2f a, v2f b, v8f c) {
    return __builtin_amdgcn_wmma_f32_16x16x4_f32(false, a, false, b, (short)0, c,
                                                 false, false);
}

// ---------------------------------------------------------------------------
// Scan kernel: grid = B/16 blocks, 320 threads (10 waves).
// Wave n computes gate columns [16n, 16n+16) for a 16-batch tile.
// ---------------------------------------------------------------------------
__global__ __launch_bounds__(320) void lstm_scan_kernel(
    const float* __restrict__ x,      // [B,S,D]
    const float* __restrict__ W_ih,   // [160,19]
    const float* __restrict__ W_hh,   // [160,40]
    const float* __restrict__ b_ih,   // [160]
    const float* __restrict__ b_hh,   // [160]
    const float* __restrict__ log_thr,
    float* __restrict__ hs,           // [B,S,H] workspace
    float* __restrict__ out)
{
    __shared__ float h_buf[16][44];   // h tile (rows = local batch)
    __shared__ float g_buf[16][164];  // gate tile

    const int tid  = threadIdx.x;
    const int wave = tid >> 5;        // 0..9 -> N tile
    const int lane = tid & 31;
    const int lm   = lane & 15;
    const int lh   = lane >> 4;       // K half select
    const int b0   = blockIdx.x * 16;

    // ---- loop-invariant B fragments (kept in VGPRs for all 2048 steps) ----
    const int ncol = wave * 16 + lm;              // gate column, 0..159
    v2f wh[10];
    #pragma unroll
    for (int kc = 0; kc < 10; ++kc) {
        const int k0 = 4 * kc + 2 * lh;           // even -> 8B aligned
        wh[kc] = *(const v2f*)(W_hh + (size_t)ncol * H_ + k0);
    }
    v2f wi[5];
    #pragma unroll
    for (int kc = 0; kc < 5; ++kc) {              // K padded 19 -> 20
        const int k0 = 4 * kc + 2 * lh;
        v2f t; t.x = (k0     < D_) ? W_ih[(size_t)ncol * D_ + k0]     : 0.0f;
               t.y = (k0 + 1 < D_) ? W_ih[(size_t)ncol * D_ + k0 + 1] : 0.0f;
        wi[kc] = t;
    }
    const float bias_n = b_ih[ncol] + b_hh[ncol];

    // branchless padded-K x loads: clamped second index + 0/1 multiplier
    int   xk1[5];
    float xm1[5];
    #pragma unroll
    for (int kc = 0; kc < 5; ++kc) {
        const int k0 = 4 * kc + 2 * lh;           // always < D_
        xk1[kc] = (k0 + 1 < D_) ? (k0 + 1) : 0;
        xm1[kc] = (k0 + 1 < D_) ? 1.0f : 0.0f;
    }
    const float* xbase = x + (size_t)(b0 + lm) * S_ * D_;  // this lane's batch row

    // ---- per-thread LSTM state: 2 of the 16x40 (b,h) elements ----
    const int e0 = tid * 2;                       // 0..638, H even -> same row
    const int ub = e0 / H_;                       // local batch row
    const int uh = e0 % H_;                       // hidden index (even)
    float c0 = 0.0f, c1 = 0.0f, h0 = 0.0f, h1 = 0.0f;

    for (int i = tid; i < 16 * 44; i += 320) ((float*)h_buf)[i] = 0.0f;
    __syncthreads();

    // prologue: load x tile for s = 0
    v2f xa[5];
    {
        const float* xr = xbase;                  // s = 0
        #pragma unroll
        for (int kc = 0; kc < 5; ++kc) {
            const int k0 = 4 * kc + 2 * lh;
            v2f t; t.x = xr[k0]; t.y = xr[xk1[kc]] * xm1[kc];
            xa[kc] = t;
        }
    }

    for (int s = 0; s < S_; ++s) {
        // prefetch x tile for s+1 (no deps -> scheduler hoists into early clause,
        // latency hidden behind wmma chain + barriers + cell update)
        v2f xn[5];
        {
            const int sn = (s + 1 < S_) ? (s + 1) : s;
            const float* xr = xbase + (size_t)sn * D_;
            #pragma unroll
            for (int kc = 0; kc < 5; ++kc) {
                const int k0 = 4 * kc + 2 * lh;
                v2f t; t.x = xr[k0]; t.y = xr[xk1[kc]] * xm1[kc];
                xn[kc] = t;
            }
        }
        // A fragments of h tile from LDS
        v2f ha[10];
        #pragma unroll
        for (int kc = 0; kc < 10; ++kc) {
            const int k0 = 4 * kc + 2 * lh;
            v2f t; t.x = h_buf[lm][k0]; t.y = h_buf[lm][k0 + 1];
            ha[kc] = t;
        }
        // gates = bias + x @ W_ih^T + h @ W_hh^T  (15 wmma)
        v8f acc;
        #pragma unroll
        for (int r = 0; r < 8; ++r) acc[r] = bias_n;
        #pragma unroll
        for (int kc = 0; kc < 5; ++kc)  acc = wmma4(xa[kc], wi[kc], acc);
        #pragma unroll
        for (int kc = 0; kc < 10; ++kc) acc = wmma4(ha[kc], wh[kc], acc);
        // C layout: vgpr r, lanes0-15 -> M=r, lanes16-31 -> M=8+r; N = lane&15
        #pragma unroll
        for (int r = 0; r < 8; ++r) g_buf[r + 8 * lh][wave * 16 + lm] = acc[r];
        __syncthreads();

        // elementwise LSTM cell update (torch gate order i,f,g,o)
        const float gi0 = g_buf[ub][uh],          gi1 = g_buf[ub][uh + 1];
        const float gf0 = g_buf[ub][H_ + uh],     gf1 = g_buf[ub][H_ + uh + 1];
        const float gg0 = g_buf[ub][2 * H_ + uh], gg1 = g_buf[ub][2 * H_ + uh + 1];
        const float go0 = g_buf[ub][3 * H_ + uh], go1 = g_buf[ub][3 * H_ + uh + 1];
        c0 = fast_sig(gf0) * c0 + fast_sig(gi0) * fast_tanh(gg0);
        c1 = fast_sig(gf1) * c1 + fast_sig(gi1) * fast_tanh(gg1);
        h0 = fast_sig(go0) * fast_tanh(c0);
        h1 = fast_sig(go1) * fast_tanh(c1);
        h_buf[ub][uh] = h0; h_buf[ub][uh + 1] = h1;
        v2f hv; hv.x = h0; hv.y = h1;
        *(v2f*)(hs + ((size_t)(b0 + ub) * S_ + s) * H_ + uh) = hv;  // 8B aligned
        __syncthreads();

        // rotate pipeline
        #pragma unroll
        for (int kc = 0; kc < 5; ++kc) xa[kc] = xn[kc];
    }

    // final states
    {
        const size_t idx = (size_t)(b0 + ub) * H_ + uh;
        v2f hv; hv.x = h0; hv.y = h1;
        v2f cv; cv.x = c0; cv.y = c1;
        *(v2f*)(out + HN_OFF + idx) = hv;
        *(v2f*)(out + CN_OFF + idx) = cv;
    }
    if (blockIdx.x == 0 && tid == 0) out[THR_OFF] = expf(log_thr[0]);
}

// ---------------------------------------------------------------------------
// Head kernel: one wave per 16-row tile of hs (rows = b*S + s).
// grid = 8192 blocks x 256 threads (8 waves) -> 65536 tiles exactly.
// ---------------------------------------------------------------------------
__global__ __launch_bounds__(256) void head_kernel(
    const float* __restrict__ hs,     // [B*S, 40]
    const float* __restrict__ ln_g,
    const float* __restrict__ ln_b,
    const float* __restrict__ W1,     // [40,40]
    const float* __restrict__ b1,
    const float* __restrict__ W2,     // [40,40]
    const float* __restrict__ b2,
    const float* __restrict__ Wo,     // [40]
    const float* __restrict__ bo,
    const float* __restrict__ log_thr,
    float* __restrict__ out)
{
    __shared__ float w1s[40][44];     // row n contiguous in K -> v2f B frags
    __shared__ float w2s[40][44];
    __shared__ float lngs[40], lnbs[40], b1s[40], b2s[40], wos[40];
    __shared__ float tbuf[8][16][44]; // h tile -> ln tile (in place)
    __shared__ float rbuf[8][16][44]; // mm1 out -> y

    const int tid = threadIdx.x;
    for (int i = tid; i < 1600; i += 256) {
        w1s[i / 40][i % 40] = W1[i];
        w2s[i / 40][i % 40] = W2[i];
    }
    if (tid < 40) {
        lngs[tid] = ln_g[tid]; lnbs[tid] = ln_b[tid];
        b1s[tid]  = b1[tid];   b2s[tid]  = b2[tid];  wos[tid] = Wo[tid];
    }
    __syncthreads();

    const int wave = tid >> 5;
    const int lane = tid & 31;
    const int lm   = lane & 15;
    const int lh   = lane >> 4;
    const float thr = expf(log_thr[0]);
    const float bo0 = bo[0];

    const size_t tile = (size_t)blockIdx.x * 8 + wave;
    const size_t r0   = tile * 16;
    float (*T)[44] = tbuf[wave];
    float (*R)[44] = rbuf[wave];

    // load 16 rows x 40 = 640 contiguous floats
    const float* src = hs + r0 * H_;
    for (int i = lane; i < 640; i += 32) T[i / 40][i % 40] = src[i];

    // LayerNorm: lanes 0..15 each own one row (in-wave DS ordering keeps this safe)
    if (lane < 16) {
        float mu = 0.0f;
        for (int c = 0; c < 40; ++c) mu += T[lane][c];
        mu *= (1.0f / 40.0f);
        float var = 0.0f;
        for (int c = 0; c < 40; ++c) { const float d = T[lane][c] - mu; var += d * d; }
        var *= (1.0f / 40.0f);
        const float rs = rsqrtf(var + LN_EPS);
        for (int c = 0; c < 40; ++c)
            T[lane][c] = (T[lane][c] - mu) * rs * lngs[c] + lnbs[c];
    }

    // ---- MM1: res = tanh(ln @ W1^T + b1) ----
    v2f a1[10];
    #pragma unroll
    for (int kc = 0; kc < 10; ++kc) {
        const int k0 = 4 * kc + 2 * lh;
        v2f t; t.x = T[lm][k0]; t.y = T[lm][k0 + 1];
        a1[kc] = t;
    }
    #pragma unroll
    for (int nt = 0; nt < 3; ++nt) {
        const int n = nt * 16 + lm;                 // output col, pad 40->48
        const float bn = (n < H_) ? b1s[n] : 0.0f;
        v8f acc;
        #pragma unroll
        for (int r = 0; r < 8; ++r) acc[r] = bn;
        #pragma unroll
        for (int kc = 0; kc < 10; ++kc) {
            const int k0 = 4 * kc + 2 * lh;
            v2f bf;
            if (n < H_) { bf.x = w1s[n][k0]; bf.y = w1s[n][k0 + 1]; }
            else        { bf.x = 0.0f;       bf.y = 0.0f; }
            acc = wmma4(a1[kc], bf, acc);
        }
        #pragma unroll
        for (int r = 0; r < 8; ++r) {
            const int M = r + 8 * lh;
            if (n < 44) R[M][n] = fast_tanh(acc[r]); // cols 40..43 get tanh(0)=0
        }
    }

    // ---- MM2 + residual: y = sigmoid(ln) + tanh(res @ W2^T + b2) ----
    v2f a2[10];
    #pragma unroll
    for (int kc = 0; kc < 10; ++kc) {
        const int k0 = 4 * kc + 2 * lh;
        v2f t; t.x = R[lm][k0]; t.y = R[lm][k0 + 1];
        a2[kc] = t;
    }
    #pragma unroll
    for (int nt = 0; nt < 3; ++nt) {
        const int n = nt * 16 + lm;
        const float bn = (n < H_) ? b2s[n] : 0.0f;
        v8f acc;
        #pragma unroll
        for (int r = 0; r < 8; ++r) acc[r] = bn;
        #pragma unroll
        for (int kc = 0; kc < 10; ++kc) {
            const int k0 = 4 * kc + 2 * lh;
            v2f bf;
            if (n < H_) { bf.x = w2s[n][k0]; bf.y = w2s[n][k0 + 1]; }
            else        { bf.x = 0.0f;       bf.y = 0.0f; }
            acc = wmma4(a2[kc], bf, acc);
        }
        #pragma unroll
        for (int r = 0; r < 8; ++r) {
            const int M = r + 8 * lh;
            if (n < H_) R[M][n] = fast_sig(T[M][n]) + fast_tanh(acc[r]); // y
        }
    }

    // ---- output head: raw = tanh(y . Wo + bo); threshold mask ----
    if (lane < 16) {
        float sdot = bo0;
        for (int c = 0; c < 40; ++c) sdot += R[lane][c] * wos[c];
        const float raw = fast_tanh(sdot);
        out[r0 + lane] = (fabsf(raw) >= thr) ? raw : 0.0f;
    }
}

// ---------------------------------------------------------------------------
extern "C" void kernel_launch(void* const* d_in, const int* in_sizes, int n_in,
                              void* d_out, int out_size, void* d_ws, size_t ws_size,
                              hipStream_t stream) {
    const float* x       = (const float*)d_in[0];
    const float* W_ih    = (const float*)d_in[1];
    const float* W_hh    = (const float*)d_in[2];
    const float* b_ih    = (const float*)d_in[3];
    const float* b_hh    = (const float*)d_in[4];
    const float* ln_g    = (const float*)d_in[5];
    const float* ln_b    = (const float*)d_in[6];
    const float* W1      = (const float*)d_in[7];
    const float* b1      = (const float*)d_in[8];
    const float* W2      = (const float*)d_in[9];
    const float* b2      = (const float*)d_in[10];
    const float* Wo      = (const float*)d_in[11];
    const float* bo      = (const float*)d_in[12];
    const float* log_thr = (const float*)d_in[13];
    float* out = (float*)d_out;
    float* hs  = (float*)d_ws;   // [B,S,H] fp32 = 168 MB

    lstm_scan_kernel<<<B_ / 16, 320, 0, stream>>>(x, W_ih, W_hh, b_ih, b_hh,
                                                  log_thr, hs, out);
    head_kernel<<<(B_ * S_) / (16 * 8), 256, 0, stream>>>(hs, ln_g, ln_b, W1, b1,
                                                          W2, b2, Wo, bo, log_thr,
                                                          out);
}